// ModelBasicCNN_58780922413605
// MI455X (gfx1250) — compile-verified
//
#include <hip/hip_runtime.h>
#include <hip/hip_bf16.h>
#include <stdint.h>

// MI455X / gfx1250, wave32. fp32 model -> V_WMMA_F32_16X16X4_F32 for the two
// large convolutions (implicit GEMM); async-to-LDS + TDM for data movement.

typedef __attribute__((ext_vector_type(2))) float v2f;
typedef __attribute__((ext_vector_type(8))) float v8f;
typedef __attribute__((ext_vector_type(4))) unsigned int u32x4;
typedef __attribute__((ext_vector_type(4))) int i32x4;
typedef __attribute__((ext_vector_type(8))) int i32x8;

// GCC-style vectors matching the async-to-LDS builtin parameter types
// (diagnostic: "__attribute__((__vector_size__(2 * sizeof(int)))) int __device__ *")
typedef int gv2i __attribute__((vector_size(8)));
typedef int gv4i __attribute__((vector_size(16)));
typedef __attribute__((address_space(1))) gv2i* glob_v2i;
typedef __attribute__((address_space(3))) gv2i* lds_v2i;
typedef __attribute__((address_space(1))) gv4i* glob_v4i;
typedef __attribute__((address_space(3))) gv4i* lds_v4i;

#ifndef __has_builtin
#define __has_builtin(x) 0
#endif

#if __has_builtin(__builtin_amdgcn_global_load_async_to_lds_b64)
#define USE_ASYNC_B64 1
#else
#define USE_ASYNC_B64 0
#endif
#if __has_builtin(__builtin_amdgcn_global_load_async_to_lds_b128)
#define USE_ASYNC_B128 1
#else
#define USE_ASYNC_B128 0
#endif
#if __has_builtin(__builtin_amdgcn_tensor_load_to_lds) && \
    __has_builtin(__builtin_amdgcn_s_wait_tensorcnt)
#define USE_TDM 1
#else
#define USE_TDM 0
#endif

__device__ __forceinline__ void wait_asynccnt0() {
#if __has_builtin(__builtin_amdgcn_s_wait_asynccnt)
  __builtin_amdgcn_s_wait_asynccnt(0);
#else
  asm volatile("s_wait_asynccnt 0" ::: "memory");
#endif
}

// ---------------------------------------------------------------------------
// conv0: 8x8, stride 2, SAME(pad 3), Cin=3, Cout=64, + b0, ReLU. Direct VALU.
// out[64,48,48,64] NHWC
// ---------------------------------------------------------------------------
__global__ __launch_bounds__(256) void conv0_kernel(
    const float* __restrict__ x, const float* __restrict__ w0,
    const float* __restrict__ b0, float* __restrict__ out)
{
  const int total = 64 * 48 * 48 * 64;
  int idx = blockIdx.x * 256 + threadIdx.x;
  if (idx >= total) return;
  int oc = idx & 63;
  int t  = idx >> 6;
  int ow = t % 48; t /= 48;
  int oh = t % 48;
  int b  = t / 48;
  float acc = b0[oc];
  for (int kh = 0; kh < 8; ++kh) {
    int ih = oh * 2 - 3 + kh;
    if ((unsigned)ih >= 96u) continue;
    for (int kw = 0; kw < 8; ++kw) {
      int iw = ow * 2 - 3 + kw;
      if ((unsigned)iw >= 96u) continue;
      const float* px = x  + ((b * 96 + ih) * 96 + iw) * 3;
      const float* pw = w0 + ((kh * 8 + kw) * 3) * 64 + oc;
      acc = fmaf(px[0], pw[0],   acc);
      acc = fmaf(px[1], pw[64],  acc);
      acc = fmaf(px[2], pw[128], acc);
    }
  }
  out[idx] = fmaxf(acc, 0.f);
}

// ---------------------------------------------------------------------------
// Implicit-GEMM conv via V_WMMA_F32_16X16X4_F32. Cout fixed = 128.
// Block = 128 threads (4 wave32). Block tile: 16 output positions x 128 chans.
// A(16x16c): per-lane GLOBAL_LOAD_ASYNC_TO_LDS_B64 (ASYNCcnt).
// B(16c x 128): one TENSOR_LOAD_TO_LDS 2D-tile DMA issued by wave 0 (TENSORcnt).
// M = B*Hout*Wout is an exact multiple of 16 -> EXEC all ones around WMMA.
// ---------------------------------------------------------------------------
__global__ __launch_bounds__(128) void conv_wmma_f32(
    const float* __restrict__ in, const float* __restrict__ w,
    const float* __restrict__ bias, float* __restrict__ out,
    int Hin, int Win, int Cin, int Hout, int Wout,
    int KH, int KW, int stride, int relu)
{
  constexpr int COUT = 128;
  // 18-column pad: even stride keeps 8B-aligned LDS dests for async b64 and
  // stays bank-conflict-free for column reads (18*L mod 64 distinct, L<16).
  __shared__ __align__(16) float Asm[16][18];
  __shared__ __align__(16) float Bsm[16][COUT];

  const int tid    = threadIdx.x;
  const int lane   = tid & 31;
  const int wave   = tid >> 5;
  const int n0     = wave * 32;
  const int m_base = blockIdx.x * 16;

  // A staging coords: 8 threads per row, 2 floats (8 bytes) per thread
  const int ar = tid >> 3;            // 0..15: local output position
  const int ac = (tid & 7) * 2;       // channel pair within chunk
  // B fallback staging coords: 8 threads per row, 16 floats per thread
  const int br = ar;
  const int bc = (tid & 7) * 16;

  // Decode this thread's staging position -> (batch, oh, ow)
  int m   = m_base + ar;
  int bN  = m / (Hout * Wout);
  int rem = m - bN * (Hout * Wout);
  int oh  = rem / Wout;
  int ow  = rem - oh * Wout;

  // WMMA fragment coords (CDNA5 ISA, 32-bit A 16x4 / C 16x16 layouts)
  const int hi = lane >> 4;           // 0: K=0,1 / M=0..7   1: K=2,3 / M=8..15
  const int lr = lane & 15;

  v8f acc0 = {0.f, 0.f, 0.f, 0.f, 0.f, 0.f, 0.f, 0.f};
  v8f acc1 = {0.f, 0.f, 0.f, 0.f, 0.f, 0.f, 0.f, 0.f};

  const int KK = KH * KW;
  for (int kk = 0; kk < KK; ++kk) {
    const int kh = kk / KW;
    const int kw = kk - kh * KW;
    const float* wslice = w + (kk * Cin) * COUT;
    if (kk + 1 < KK) {
      // warm caches for the next weight slice (global_prefetch_b8)
      __builtin_prefetch(wslice + Cin * COUT + tid * 16, 0, 1);
    }
    const int ih = oh * stride + kh;   // VALID conv: always in bounds
    const int iw = ow * stride + kw;
    const float* arow = in + ((bN * Hin + ih) * Win + iw) * Cin;

    for (int c0 = 0; c0 < Cin; c0 += 16) {
      __syncthreads();   // previous fragment reads done before restaging

      // ---- Stage A: 16 positions x 16 input channels (NHWC contiguous) ----
#if USE_ASYNC_B64
      __builtin_amdgcn_global_load_async_to_lds_b64(
          (glob_v2i)(arow + c0 + ac),
          (lds_v2i)&Asm[ar][ac], 0, 0);
#else
      {
        float2 av = *(const float2*)(arow + c0 + ac);
        Asm[ar][ac]     = av.x;
        Asm[ar][ac + 1] = av.y;
      }
#endif

      // ---- Stage B: 16 input channels x 128 output channels (HWIO) ----
#if USE_TDM
      if (wave == 0) {
        // Tensor DMA descriptor (CDNA5 ISA):
        // 2D tile: tile_dim0=128 f32 (one weight row), tile_dim1=16 rows,
        // tensor_dim0_stride=128 -> contiguous 8 KB into Bsm.
        unsigned long long ga =
            (unsigned long long)(uintptr_t)(wslice + c0 * COUT);
        unsigned lds_off = (unsigned)(uintptr_t)&Bsm[0][0]; // generic low32 = LDS addr
        u32x4 g0 = { 1u,                                  // count=1 (valid D#)
                     lds_off,                              // lds_addr
                     (unsigned)ga,                         // global_addr[31:0]
                     (unsigned)(ga >> 32) | (2u << 30) };  // ga[56:32] | type=2
        i32x8 g1 = { (int)(2u << 16),     // workgroup_mask=0, data_size=4B
                     (int)(128u << 16),   // tensor_dim0 = 128 (low16)
                     (int)(16u << 16),    // tensor_dim1 = 16 (low16)
                     (int)(128u << 16),   // tile_dim0 = 128
                     16,                  // tile_dim1 = 16, tile_dim2 = 0
                     128,                 // tensor_dim0_stride = 128
                     0, 0 };              // tensor_dim1_stride = 0
        i32x4 z4 = { 0, 0, 0, 0 };
#if __clang_major__ >= 23
        i32x8 z8 = { 0, 0, 0, 0, 0, 0, 0, 0 };
        __builtin_amdgcn_tensor_load_to_lds(g0, g1, z4, z4, z8, 0);
#else
        __builtin_amdgcn_tensor_load_to_lds(g0, g1, z4, z4, 0);
#endif
        __builtin_amdgcn_s_wait_tensorcnt(0);
      }
#else
      {
        const float* wrow = wslice + (c0 + br) * COUT + bc;
        float4 wv0 = *(const float4*)(wrow + 0);
        float4 wv1 = *(const float4*)(wrow + 4);
        float4 wv2 = *(const float4*)(wrow + 8);
        float4 wv3 = *(const float4*)(wrow + 12);
        *(float4*)(&Bsm[br][bc + 0])  = wv0;
        *(float4*)(&Bsm[br][bc + 4])  = wv1;
        *(float4*)(&Bsm[br][bc + 8])  = wv2;
        *(float4*)(&Bsm[br][bc + 12]) = wv3;
      }
#endif

#if USE_ASYNC_B64
      wait_asynccnt0();
#endif
      __syncthreads();

      #pragma unroll
      for (int ks = 0; ks < 4; ++ks) {
        const int k0 = ks * 4 + 2 * hi;
        v2f a, bA, bB;
        // A 16x4 f32: lanes 0-15 hold K=0(+1), lanes 16-31 hold K=2(+3)
        a.x = Asm[lr][k0];
        a.y = Asm[lr][k0 + 1];
        // B 4x16 f32: rows striped across lanes, mirrored K split
        bA.x = Bsm[k0][n0 + lr];
        bA.y = Bsm[k0 + 1][n0 + lr];
        bB.x = Bsm[k0][n0 + 16 + lr];
        bB.y = Bsm[k0 + 1][n0 + 16 + lr];
        // 8 args: (neg_a, A, neg_b, B, c_mod, C, reuse_a, reuse_b)
        acc0 = __builtin_amdgcn_wmma_f32_16x16x4_f32(
            false, a, false, bA, (short)0, acc0, false, false);
        acc1 = __builtin_amdgcn_wmma_f32_16x16x4_f32(
            false, a, false, bB, (short)0, acc1, false, false);
      }
    }
  }

  // Epilogue: C/D 16x16 f32 layout -> VGPR r: M=r (lanes 0-15) / M=r+8 (16-31)
  #pragma unroll
  for (int t = 0; t < 2; ++t) {
    const int ncol = n0 + 16 * t + lr;
    const float bv = bias ? bias[ncol] : 0.f;
    v8f acc = t ? acc1 : acc0;
    #pragma unroll
    for (int r = 0; r < 8; ++r) {
      const int mrow = m_base + r + 8 * hi;
      float v = acc[r] + bv;
      if (relu) v = fmaxf(v, 0.f);
      out[mrow * COUT + ncol] = v;
    }
  }
}

// ---------------------------------------------------------------------------
// Hard-concrete gate samples: z[s,c] = clip(sigmoid((logit(u)+la)/0.5)*1.2-0.1)
// ---------------------------------------------------------------------------
__global__ __launch_bounds__(128) void gates_kernel(
    const float* __restrict__ u, const float* __restrict__ la,
    float* __restrict__ z)
{
  int s = blockIdx.x;       // 0..256
  int c = threadIdx.x;      // 0..127
  float uv = u[s * 128 + c];
  float x  = (logf(uv) - log1pf(-uv) + la[c]) * 2.0f;   // / TEMP(0.5)
  float sg = 1.f / (1.f + __expf(-x));
  float zz = sg * 1.2f - 0.1f;                          // * (ZETA-GAMMA) + GAMMA
  z[s * 128 + c] = fminf(fmaxf(zz, 0.f), 1.f);
}

// ---------------------------------------------------------------------------
// 257-sample scan: out[e] = (1/256) * sum_s relu(y1[e]*z[s,c] + b1[c])
// 4 elements per thread (all share channel c = tid&127); z chunks streamed
// into LDS via GLOBAL_LOAD_ASYNC_TO_LDS_B128 when available.
// ---------------------------------------------------------------------------
__global__ __launch_bounds__(256) void scan_kernel(
    const float* __restrict__ y1, const float* __restrict__ z,
    const float* __restrict__ b1, float* __restrict__ out)
{
  __shared__ __align__(16) float zsm[64 * 128];   // 32 KB
  const int tid = threadIdx.x;
  const int base = blockIdx.x * 1024 + tid;
  const int c = tid & 127;
  const float bv = b1[c];
  float y[4], acc[4];
  #pragma unroll
  for (int j = 0; j < 4; ++j) { y[j] = y1[base + 256 * j]; acc[j] = 0.f; }

  for (int s0 = 0; s0 < 257; s0 += 64) {
    const int len = (257 - s0) < 64 ? (257 - s0) : 64;
    __syncthreads();
#if USE_ASYNC_B128
    for (int off = tid * 4; off < len * 128; off += 256 * 4) {
      __builtin_amdgcn_global_load_async_to_lds_b128(
          (glob_v4i)(z + s0 * 128 + off),
          (lds_v4i)&zsm[off], 0, 0);
    }
    wait_asynccnt0();
#else
    for (int i = tid; i < len * 128; i += 256) zsm[i] = z[s0 * 128 + i];
#endif
    __syncthreads();
    for (int s = 0; s < len; ++s) {
      const float zc = zsm[s * 128 + c];
      #pragma unroll
      for (int j = 0; j < 4; ++j)
        acc[j] += fmaxf(fmaf(y[j], zc, bv), 0.f);
    }
  }
  const float inv = 1.f / 256.f;    // reference divides by L, not L+1
  #pragma unroll
  for (int j = 0; j < 4; ++j) out[base + 256 * j] = acc[j] * inv;
}

// ---------------------------------------------------------------------------
// Dense: logits[b,j] = sum_k act[b,k] * wd[k,j] + bd[j]   (K=41472, N=10)
// One block per batch row; wd is 1.66 MB -> stays L2-resident.
// ---------------------------------------------------------------------------
__global__ __launch_bounds__(256) void dense_kernel(
    const float* __restrict__ act, const float* __restrict__ wd,
    const float* __restrict__ bd, float* __restrict__ out)
{
  __shared__ float red[10][256];
  const int b = blockIdx.x, t = threadIdx.x;
  const float* row = act + b * 41472;
  float acc[10];
  #pragma unroll
  for (int j = 0; j < 10; ++j) acc[j] = 0.f;
  for (int k = t; k < 41472; k += 256) {
    const float a = row[k];
    const float* wr = wd + k * 10;
    #pragma unroll
    for (int j = 0; j < 10; ++j) acc[j] = fmaf(a, wr[j], acc[j]);
  }
  #pragma unroll
  for (int j = 0; j < 10; ++j) red[j][t] = acc[j];
  __syncthreads();
  for (int off = 128; off > 0; off >>= 1) {
    if (t < off) {
      #pragma unroll
      for (int j = 0; j < 10; ++j) red[j][t] += red[j][t + off];
    }
    __syncthreads();
  }
  if (t < 10) out[b * 10 + t] = red[t][0] + bd[t];
}

// ---------------------------------------------------------------------------
// penalty = sum_c sigmoid(log_alpha[c] + 0.5*log(11))
// ---------------------------------------------------------------------------
__global__ __launch_bounds__(128) void penalty_kernel(
    const float* __restrict__ la, float* __restrict__ out_scalar)
{
  __shared__ float red[128];
  const int c = threadIdx.x;
  const float x = la[c] + 1.1989476363991853f;
  red[c] = 1.f / (1.f + __expf(-x));
  __syncthreads();
  for (int off = 64; off > 0; off >>= 1) {
    if (c < off) red[c] += red[c + off];
    __syncthreads();
  }
  if (c == 0) out_scalar[0] = red[0];
}

// ---------------------------------------------------------------------------
extern "C" void kernel_launch(void* const* d_in, const int* in_sizes, int n_in,
                              void* d_out, int out_size, void* d_ws, size_t ws_size,
                              hipStream_t stream) {
  const float* x  = (const float*)d_in[0];   // [64,96,96,3]
  const float* w0 = (const float*)d_in[1];   // [8,8,3,64]
  const float* b0 = (const float*)d_in[2];   // [64]
  const float* w1 = (const float*)d_in[3];   // [6,6,64,128]
  const float* b1 = (const float*)d_in[4];   // [128]
  const float* la = (const float*)d_in[5];   // [128]
  const float* u  = (const float*)d_in[6];   // [257,128]
  const float* w2 = (const float*)d_in[7];   // [5,5,128,128]
  const float* b2 = (const float*)d_in[8];   // [128]
  const float* wd = (const float*)d_in[9];   // [41472,10]
  const float* bd = (const float*)d_in[10];  // [10]
  float* out = (float*)d_out;                // 640 logits + 1 penalty

  // Workspace layout (floats): total ~80.2 MB, L2-resident (192 MB L2)
  float* net0 = (float*)d_ws;                // 64*48*48*64  = 9437184
  float* y1   = net0 + 9437184;              // 64*22*22*128 = 3964928
  float* net1 = y1   + 3964928;              // same
  float* act2 = net1 + 3964928;              // 64*18*18*128 = 2654208
  float* z    = act2 + 2654208;              // 257*128      = 32896

  // conv0: [64,96,96,3] -> [64,48,48,64]
  conv0_kernel<<<(64 * 48 * 48 * 64) / 256, 256, 0, stream>>>(x, w0, b0, net0);

  // gate samples z[257,128]
  gates_kernel<<<257, 128, 0, stream>>>(u, la, z);

  // conv1 (WMMA): M = 64*22*22 = 30976, K = 6*6*64, no bias / no relu
  conv_wmma_f32<<<30976 / 16, 128, 0, stream>>>(
      net0, w1, nullptr, y1, 48, 48, 64, 22, 22, 6, 6, 2, 0);

  // 257-sample hard-concrete scan
  scan_kernel<<<3964928 / 1024, 256, 0, stream>>>(y1, z, b1, net1);

  // conv2 (WMMA): M = 64*18*18 = 20736, K = 5*5*128, + b2, relu
  conv_wmma_f32<<<20736 / 16, 128, 0, stream>>>(
      net1, w2, b2, act2, 22, 22, 128, 18, 18, 5, 5, 1, 1);

  // dense + penalty
  dense_kernel<<<64, 256, 0, stream>>>(act2, wd, bd, out);
  penalty_kernel<<<1, 128, 0, stream>>>(la, out + 640);
}